// SpatialTransformer_69827578298941
// MI455X (gfx1250) — compile-verified
//
#include <hip/hip_runtime.h>

typedef __attribute__((ext_vector_type(2))) float v2f;
typedef __attribute__((ext_vector_type(8))) float v8f;

#define D0 160
#define D1 192
#define D2 160
#define NV (D0 * D1 * D2)   // 4,915,200 voxels per batch (divisible by 256)

__global__ __launch_bounds__(256)
void st_affine_trilinear_kernel(const float* __restrict__ vol,
                                const float* __restrict__ trf,
                                float* __restrict__ out)
{
    const int b    = blockIdx.z;
    const int vox  = blockIdx.x * blockDim.x + threadIdx.x;  // voxel id within batch
    const int lane = threadIdx.x & 31;                        // wave32
    const int waveBase = vox - lane;
    const int n    = lane & 15;
    const bool hi  = lane >= 16;

    // Affine matrix (uniform per batch) in scalar registers:
    // M = trf.reshape(3,4) + I[:3,:4]
    const float* t = trf + b * 12;
    const float m00 = t[0] + 1.0f, m01 = t[1],        m02 = t[2],         m03 = t[3];
    const float m10 = t[4],        m11 = t[5] + 1.0f, m12 = t[6],         m13 = t[7];
    const float m20 = t[8],        m21 = t[9],        m22 = t[10] + 1.0f, m23 = t[11];

    const float c0 = (D0 - 1) * 0.5f;
    const float c1 = (D1 - 1) * 0.5f;
    const float c2 = (D2 - 1) * 0.5f;

    // ---- A fragment: 16x4 f32 (2 VGPRs/lane).
    // Lanes 0-15: M=lane, v0=K0, v1=K1.  Lanes 16-31: M=lane-16, v0=K2, v1=K3.
    // Affine rows live in matrix rows 0-2 AND 8-10 (so high lanes of D carry coords too).
    // Built with pure register selects (no memory-backed array -> no DS/scratch traffic).
    const int r = (n < 3) ? n : ((n >= 8 && n < 11) ? (n - 8) : -1);
    const float a0r0 = hi ? m02 : m00;   // row 0, K = k0
    const float a0r1 = hi ? m12 : m10;
    const float a0r2 = hi ? m22 : m20;
    const float a1r0 = hi ? m03 : m01;   // row 0, K = k1
    const float a1r1 = hi ? m13 : m11;
    const float a1r2 = hi ? m23 : m21;
    v2f afrag;
    afrag.x = (r == 0) ? a0r0 : (r == 1) ? a0r1 : (r == 2) ? a0r2 : 0.0f;
    afrag.y = (r == 0) ? a1r0 : (r == 1) ? a1r1 : (r == 2) ? a1r2 : 0.0f;

    // ---- B fragments: 4x16 f32, column n = centered homogeneous coords of a voxel.
    // v0: lanes0-15 K=0 (xc0), lanes16-31 K=2 (xc2); v1: K=1 (xc1) / K=3 (1.0)
    auto centered = [&](int v, float& xc0, float& xc1, float& xc2) {
        int x2 = v % D2;
        int x1 = (v / D2) % D1;
        int x0 = v / (D1 * D2);
        xc0 = (float)x0 - c0;
        xc1 = (float)x1 - c1;
        xc2 = (float)x2 - c2;
    };
    float p0, p1, p2, q0, q1, q2;
    centered(waveBase + n,      p0, p1, p2);   // columns for lanes 0-15's voxels
    centered(waveBase + 16 + n, q0, q1, q2);   // columns for lanes 16-31's voxels
    v2f bf1, bf2;
    bf1.x = hi ? p2 : p0;  bf1.y = hi ? 1.0f : p1;
    bf2.x = hi ? q2 : q0;  bf2.y = hi ? 1.0f : q1;

    // D = A x B (+0): rows 0-2 (low lanes) / 8-10 (high lanes) = sample coords.
    v8f cz = {};
    v8f d1 = __builtin_amdgcn_wmma_f32_16x16x4_f32(false, afrag, false, bf1,
                                                   (short)0, cz, false, false);
    v8f d2 = __builtin_amdgcn_wmma_f32_16x16x4_f32(false, afrag, false, bf2,
                                                   (short)0, cz, false, false);

    // Every lane picks its own voxel's coords (lane<16 -> d1 rows 0-2,
    // lane>=16 -> d2 rows 8-10, both at fragment elements 0..2). Add back centers.
    const float s0 = (hi ? d2[0] : d1[0]) + c0;
    const float s1 = (hi ? d2[1] : d1[1]) + c1;
    const float s2 = (hi ? d2[2] : d1[2]) + c2;

    // ---- Trilinear interp, replicating reference clip/floor/weight semantics.
    auto dimidx = [](float f, float mx, int& i0, int& i1, float& w0) {
        float cl = fminf(fmaxf(f, 0.0f), mx);
        float f0 = fminf(fmaxf(floorf(f), 0.0f), mx);
        float f1 = fminf(f0 + 1.0f, mx);
        i0 = (int)f0;
        i1 = (int)f1;
        w0 = f1 - cl;           // weight of corner i0; corner i1 gets 1 - w0
    };
    int   i00, i01, i10, i11, i20, i21;
    float w00, w10, w20;
    dimidx(s0, (float)(D0 - 1), i00, i01, w00);
    dimidx(s1, (float)(D1 - 1), i10, i11, w10);
    dimidx(s2, (float)(D2 - 1), i20, i21, w20);
    const float w01 = 1.0f - w00;
    const float w11 = 1.0f - w10;
    const float w21 = 1.0f - w20;

    const float* __restrict__ vb = vol + (size_t)b * NV;
    const int r0a = i00 * (D1 * D2), r0b = i01 * (D1 * D2);
    const int r1a = i10 * D2,        r1b = i11 * D2;

    float acc;
    acc  = (w00 * w10) * (w20 * vb[r0a + r1a + i20] + w21 * vb[r0a + r1a + i21]);
    acc += (w00 * w11) * (w20 * vb[r0a + r1b + i20] + w21 * vb[r0a + r1b + i21]);
    acc += (w01 * w10) * (w20 * vb[r0b + r1a + i20] + w21 * vb[r0b + r1a + i21]);
    acc += (w01 * w11) * (w20 * vb[r0b + r1b + i20] + w21 * vb[r0b + r1b + i21]);

    out[(size_t)b * NV + vox] = acc;
}

extern "C" void kernel_launch(void* const* d_in, const int* in_sizes, int n_in,
                              void* d_out, int out_size, void* d_ws, size_t ws_size,
                              hipStream_t stream) {
    (void)in_sizes; (void)n_in; (void)d_ws; (void)ws_size; (void)out_size;
    const float* vol = (const float*)d_in[0];   // (2,160,192,160,1) f32
    const float* trf = (const float*)d_in[1];   // (2,12) f32
    float* out = (float*)d_out;                 // (2,160,192,160,1) f32

    dim3 grid(NV / 256, 1, 2);                  // 19200 blocks per batch, z = batch
    st_affine_trilinear_kernel<<<grid, dim3(256, 1, 1), 0, stream>>>(vol, trf, out);
}